// QCNN_65481071396400
// MI455X (gfx1250) — compile-verified
//
#include <hip/hip_runtime.h>
#include <stdint.h>

// ---------------------------------------------------------------------------
// MLP 8->16->16->12->8->4->4->1 with BN(eval)+ReLU fused into the linears,
// computed with v_wmma_f16_16x16x32_f16 on gfx1250 (wave32).
//
// Transposed-activation scheme:  Y^T = Wfused * X^T  (+ bias via C operand).
//   A operand = fused weights (M=dout<=16, K=din<=16, zero padded; K>=16 zero)
//   B operand = activations (K=din, N=16 batch rows)
//   C/D      = f16 16x16: lane=N, u[r]=(M=2r lo, M=2r+1 hi), lanes16-31 M+=8
//
// KEY: the 16-bit C/D layout is IDENTICAL to the f16 B-operand packing
// (u[v<4] = K=2v lo, K=2v+1 hi, lanes16-31 K+=8), so layer chaining needs only:
//   B.u[j]   = v_pk_max_num_f16(D.u[j], 0)        (packed ReLU, both halves)
//   B.u[4+j] = ds_swizzle SWAPX16(B.u[j])         (K=16..31, zero A cols; only if dout>8)
// Stale/garbage B dwords multiply zero-padded A columns (0*finite==0), so no
// masking is needed.
// ---------------------------------------------------------------------------

typedef __attribute__((ext_vector_type(16))) _Float16 v16h;
typedef __attribute__((ext_vector_type(8)))  _Float16 v8h;

union Tile16 { v16h h; uint32_t u[8]; };   // A or B operand (8 VGPRs)
union Acc16  { v8h  h; uint32_t u[4]; };   // f16 C/D operand (4 VGPRs)

static __device__ __forceinline__ uint32_t pk16(float a, float b) {
    auto p = __builtin_amdgcn_cvt_pkrtz(a, b);   // v_cvt_pk_rtz_f16_f32
    return __builtin_bit_cast(uint32_t, p);
}
static __device__ __forceinline__ uint32_t swap16(uint32_t v) {
    // ds_swizzle group-of-32, xor_mask=0x10, and_mask=0x1f  (SWAPX16)
    return (uint32_t)__builtin_amdgcn_ds_swizzle((int)v, 0x401f);
}
static __device__ __forceinline__ uint32_t pkrelu(uint32_t v) {
    uint32_t r;
    asm("v_pk_max_num_f16 %0, %1, 0" : "=v"(r) : "v"(v));  // packed ReLU
    return r;
}
static __device__ __forceinline__ void unpk2(uint32_t v, float& a, float& b) {
    union { uint32_t u; _Float16 h[2]; } c; c.u = v;
    a = (float)c.h[0]; b = (float)c.h[1];
}

struct ParamPtrs { const float* p[40]; };

// ---------------------------------------------------------------------------
// Prologue: fold BN into the linears.
// wsW : [layer][m(16)][k(16)] f16 zero-padded (A tiles)
// wsBp: [layer][8] u32 packed-f16 bias in C/D layout:
//       wsBp[i*8 + half*4 + r] = (bf[2r+8*half] lo, bf[2r+1+8*half] hi)
// wsF : [w0,w1,w2,w3,bias] f32 (final 4->1 linear)
// ---------------------------------------------------------------------------
__global__ __launch_bounds__(256) void fuse_params(ParamPtrs P,
                                                   unsigned short* wsW,
                                                   uint32_t* wsBp, float* wsF)
{
    constexpr int DIMS[7] = {8, 16, 16, 12, 8, 4, 4};
    const int t = threadIdx.x;      // 256 threads = one 16x16 tile
    const int m = t >> 4;
    const int k = t & 15;
#pragma unroll
    for (int i = 0; i < 6; ++i) {
        const float* W     = P.p[1 + 6 * i + 0];
        const float* b     = P.p[1 + 6 * i + 1];
        const float* gamma = P.p[1 + 6 * i + 2];
        const float* beta  = P.p[1 + 6 * i + 3];
        const float* rmean = P.p[1 + 6 * i + 4];
        const float* rvar  = P.p[1 + 6 * i + 5];
        const int din = DIMS[i], dout = DIMS[i + 1];
        float scale = 0.0f, wv = 0.0f;
        if (m < dout) scale = gamma[m] * rsqrtf(rvar[m] + 1e-5f);
        if (m < dout && k < din) wv = W[m * din + k] * scale;
        _Float16 hv = (_Float16)wv;
        wsW[i * 256 + m * 16 + k] = __builtin_bit_cast(unsigned short, hv);
        // packed bias: threads 0..7 each build one dword (pair of features)
        if (t < 8) {
            const int m0 = 2 * t;            // t=0..3 -> M 0..7 ; t=4..7 -> 8..15
            auto bfuse = [&](int mm) -> float {
                if (mm >= dout) return 0.0f;
                float sc = gamma[mm] * rsqrtf(rvar[mm] + 1e-5f);
                return sc * (b[mm] - rmean[mm]) + beta[mm];
            };
            wsBp[i * 8 + t] = pk16(bfuse(m0), bfuse(m0 + 1));
        }
    }
    if (t < 4)  wsF[t] = P.p[37][t];
    if (t == 4) wsF[4] = P.p[38][0];
}

// ---------------------------------------------------------------------------
// Main kernel: each wave handles 32 rows per trip (two 16-row WMMA chains),
// grid-stride over row tiles.
// ---------------------------------------------------------------------------
__global__ __launch_bounds__(256) void mlp_forward(
    const float* __restrict__ x, const unsigned short* __restrict__ wsWu,
    const uint32_t* __restrict__ wsBp, const float* __restrict__ wsF,
    float* __restrict__ out, int nrows)
{
    constexpr int DIMS[7] = {8, 16, 16, 12, 8, 4, 4};
    const int lane  = threadIdx.x & 31;
    const int half  = lane >> 4;   // K-half for A, M-half for C/D
    const int row16 = lane & 15;

    // ---- per-wave constants: A tiles and packed-f16 bias C tiles ---------
    Tile16 A[6];
    Acc16  biasC[6];
#pragma unroll
    for (int i = 0; i < 6; ++i) {
        const uint4* wp =
            (const uint4*)(wsWu + i * 256 + row16 * 16 + half * 8);
        uint4 w = *wp;
        A[i].u[0] = w.x; A[i].u[1] = w.y; A[i].u[2] = w.z; A[i].u[3] = w.w;
        A[i].u[4] = 0u;  A[i].u[5] = 0u;  A[i].u[6] = 0u;  A[i].u[7] = 0u;
        const uint4* bp = (const uint4*)(wsBp + i * 8 + half * 4);
        uint4 bb = *bp;
        biasC[i].u[0] = bb.x; biasC[i].u[1] = bb.y;
        biasC[i].u[2] = bb.z; biasC[i].u[3] = bb.w;
    }
    const float fw0 = wsF[0], fw1 = wsF[1], fw2 = wsF[2], fw3 = wsF[3];
    const float fbv = wsF[4];

    const int  wavesPerBlock = blockDim.x >> 5;
    const long wid    = (long)blockIdx.x * wavesPerBlock + (threadIdx.x >> 5);
    const long nWaves = (long)gridDim.x * wavesPerBlock;
    const int  nPairs = (nrows + 31) >> 5;

    for (long tp = wid; tp < nPairs; tp += nWaves) {
        const int base = (int)tp * 32;
        const int row  = base + lane;

        // ---- load one input row per lane (8 f32 = 2x b128) ----
        float x0 = 0.f, x1 = 0.f, x2 = 0.f, x3 = 0.f;
        float x4 = 0.f, x5 = 0.f, x6 = 0.f, x7 = 0.f;
        if (row < nrows) {
            const float4* xp = (const float4*)(x + (size_t)row * 8);
            float4 a = xp[0], b = xp[1];
            x0 = a.x; x1 = a.y; x2 = a.z; x3 = a.w;
            x4 = b.x; x5 = b.y; x6 = b.z; x7 = b.w;
        }
        uint32_t p0 = pk16(x0, x1), p1 = pk16(x2, x3);
        uint32_t p2 = pk16(x4, x5), p3 = pk16(x6, x7);

        // Chain 0: rows base+0..15 (lanes 0-15 own their data).
        // Chain 1: rows base+16..31 (pull across the half-wave boundary).
        Tile16 B0, B1;
        B0.u[0] = p0; B0.u[1] = p1; B0.u[2] = p2; B0.u[3] = p3;
        B0.u[4] = 0u; B0.u[5] = 0u; B0.u[6] = 0u; B0.u[7] = 0u;
        B1.u[0] = swap16(p0); B1.u[1] = swap16(p1);
        B1.u[2] = swap16(p2); B1.u[3] = swap16(p3);
        B1.u[4] = 0u; B1.u[5] = 0u; B1.u[6] = 0u; B1.u[7] = 0u;

        Acc16 acc0, acc1;
#pragma unroll
        for (int i = 0; i < 6; ++i) {
            const int dout = DIMS[i + 1];          // == din of layer i+1
            acc0.h = __builtin_amdgcn_wmma_f16_16x16x32_f16(
                false, A[i].h, false, B0.h, (short)0, biasC[i].h, false, false);
            acc1.h = __builtin_amdgcn_wmma_f16_16x16x32_f16(
                false, A[i].h, false, B1.h, (short)0, biasC[i].h, false, false);
            if (i < 5) {
                // packed ReLU; u[j] covers M=2j,2j+1 (+8 in upper lanes) which
                // is exactly K=2j,2j+1 (+8) of the next layer's B operand.
                const int nm = dout < 8 ? dout : 8;       // live features/half
                const int npk = (nm + 1) >> 1;            // dwords to ReLU
#pragma unroll
                for (int j = 0; j < 4; ++j) {
                    if (j < npk) {
                        B0.u[j] = pkrelu(acc0.u[j]);
                        B1.u[j] = pkrelu(acc1.u[j]);
                    }
                }
                if (dout > 8) {                    // B.u[4..7] = K=16..31, zero A cols
#pragma unroll
                    for (int j = 0; j < 4; ++j) {
                        B0.u[4 + j] = swap16(B0.u[j]);
                        B1.u[4 + j] = swap16(B1.u[j]);
                    }
                }
            }
        }

        // ---- last hidden layer (dout=4): ReLU + unpack M=0..3, lanes 0-15 -
        uint32_t a01_0 = pkrelu(acc0.u[0]), a23_0 = pkrelu(acc0.u[1]);
        uint32_t a01_1 = pkrelu(acc1.u[0]), a23_1 = pkrelu(acc1.u[1]);
        float h0, h1, h2, h3, g0, g1, g2, g3;
        unpk2(a01_0, h0, h1); unpk2(a23_0, h2, h3);
        unpk2(a01_1, g0, g1); unpk2(a23_1, g2, g3);

        // ---- final Linear(4,1) + sigmoid ----
        float z0 = fbv, z1 = fbv;
        z0 = fmaf(fw0, h0, z0); z0 = fmaf(fw1, h1, z0);
        z0 = fmaf(fw2, h2, z0); z0 = fmaf(fw3, h3, z0);
        z1 = fmaf(fw0, g0, z1); z1 = fmaf(fw1, g1, z1);
        z1 = fmaf(fw2, g2, z1); z1 = fmaf(fw3, g3, z1);
        const float L2E = 1.44269504088896340736f;
        float e0 = __builtin_amdgcn_exp2f(-z0 * L2E);
        float e1 = __builtin_amdgcn_exp2f(-z1 * L2E);
        float s0 = __builtin_amdgcn_rcpf(1.0f + e0);
        float s1 = __builtin_amdgcn_rcpf(1.0f + e1);

        // Move chain-1 results (valid in lanes 0-15) to lanes 16-31, then do
        // one fully coalesced 32-lane store.
        float s1sw = __builtin_bit_cast(
            float, swap16(__builtin_bit_cast(uint32_t, s1)));
        float sv = (lane < 16) ? s0 : s1sw;
        if (row < nrows) out[row] = sv;
    }
}

// ---------------------------------------------------------------------------
extern "C" void kernel_launch(void* const* d_in, const int* in_sizes, int n_in,
                              void* d_out, int out_size, void* d_ws, size_t ws_size,
                              hipStream_t stream)
{
    (void)in_sizes; (void)ws_size;
    ParamPtrs P{};
    for (int i = 0; i < n_in && i < 40; ++i) P.p[i] = (const float*)d_in[i];

    // workspace: 6*256 f16 weights | 6*8 u32 packed bias | 5 f32 final
    unsigned short* wsW = (unsigned short*)d_ws;
    uint32_t* wsBp = (uint32_t*)((char*)d_ws + 6 * 256 * sizeof(unsigned short));
    float* wsF = (float*)((char*)wsBp + 6 * 8 * sizeof(uint32_t));

    fuse_params<<<1, 256, 0, stream>>>(P, wsW, wsBp, wsF);

    const int nrows  = out_size;           // B = 2^21 rows, one output each
    const int nPairs = (nrows + 31) >> 5;  // 32 rows per wave per trip
    const int wavesPerBlock = 8;           // 256 threads
    long blocksL = ((long)nPairs + wavesPerBlock - 1) / wavesPerBlock;
    if (blocksL < 1) blocksL = 1;
    if (blocksL > 2048) blocksL = 2048;    // persistent-ish, grid-stride

    mlp_forward<<<(int)blocksL, 256, 0, stream>>>(
        (const float*)d_in[0], wsW, wsBp, wsF, (float*)d_out, nrows);
}